// RoutingStorage_76914274337426
// MI455X (gfx1250) — compile-verified
//
#include <hip/hip_runtime.h>

typedef __attribute__((ext_vector_type(2))) float v2f;
typedef __attribute__((ext_vector_type(8))) float v8f;

// ---------------------------------------------------------------------------
// GR4J unit-hydrograph ordinates for X4 = 2.5 (static, from the two S-curves):
//   ord1 = [0.101192885, 0.471240517, 0.427566598]
//   ord2 = [0.050596443, 0.235620258, 0.427566598, 0.235620258, 0.050596443, 0]
// Folded with the 0.9 / 0.1 routing split -> FIR taps h1 (3 taps), h2 (6 taps).
// Banded matrix: H[m,k] = h[m+5-k], nonzero only for 0 <= m+5-k <= 5.
// ---------------------------------------------------------------------------
constexpr float cband1(int j) {
  return j == 0 ? 0.09107359665f
       : j == 1 ? 0.42411646530f
       : j == 2 ? 0.38480993820f
       : 0.0f;
}
constexpr float cband2(int j) {
  return j == 0 ? 0.00505964425f
       : j == 1 ? 0.02356202580f
       : j == 2 ? 0.04275665980f
       : j == 3 ? 0.02356202580f
       : j == 4 ? 0.00505964425f
       : 0.0f;  // h2[5] == 0 exactly
}

// Precomputed per-lane A-matrix fragments for V_WMMA_F32_16X16X4_F32.
// ISA layout (16x4 f32 A): lanes 0-15 hold M=0..15 with K = {4c, 4c+1} in
// comps {0,1}; lanes 16-31 hold the same M rows with K = {4c+2, 4c+3}.
// Packed per (chunk, lane) as float4 {a1.c0, a1.c1, a2.c0, a2.c1}.
struct alignas(16) ATab { float v[6][32][4]; };

constexpr ATab make_atab() {
  ATab t{};
  for (int c = 0; c < 6; ++c)
    for (int lane = 0; lane < 32; ++lane) {
      int m  = lane & 15;
      int hi = lane >> 4;
      int kA = 4 * c + 2 * hi;
      t.v[c][lane][0] = cband1(m + 5 - kA);
      t.v[c][lane][1] = cband1(m + 4 - kA);
      t.v[c][lane][2] = cband2(m + 5 - kA);
      t.v[c][lane][3] = cband2(m + 4 - kA);
    }
  return t;
}
__constant__ ATab A_TAB = make_atab();

// ---------------------------------------------------------------------------
// Kernel 1: both FIRs as banded WMMA matmuls.
// Block = 256 threads (8 wave32). Each wave produces one 256-step tile of
// u1 and u2 via 6 chained V_WMMA_F32_16X16X4_F32 per filter:
//   U[m,n] = sum_k H[m,k] * P[k,n],  P[k,n] = p_r[tile + 16n - 5 + k]
// giving u[tile + 16n + m]. p_r staged in LDS with a 5-left / 3-right halo.
// The P operand is shared by both filters (pr1 = 9*pr2), so each ds_load_b64
// feeds two WMMAs.
// ---------------------------------------------------------------------------
__global__ void __launch_bounds__(256)
fir_wmma_kernel(const float* __restrict__ x, float* __restrict__ u1,
                float* __restrict__ u2, int T) {
  __shared__ float sp[2056];  // p_r over [blockbase-5, blockbase+2050]
  const int tid = threadIdx.x;
  const int blockbase = blockIdx.x * 2048;
  const float4* __restrict__ xr = (const float4*)x;

  // Stage p_r = x.w + x.x - x.z  (perc + p_n - p_s)
#pragma unroll
  for (int j = 0; j < 8; ++j) {
    int t = blockbase + j * 256 + tid;
    float v = 0.0f;
    if (t < T) { float4 q = xr[t]; v = q.w + q.x - q.z; }
    sp[5 + j * 256 + tid] = v;
  }
  if (tid < 5) {  // left halo (zeros before t=0: zero-initialized UH state)
    int t = blockbase - 5 + tid;
    float v = 0.0f;
    if (t >= 0) { float4 q = xr[t]; v = q.w + q.x - q.z; }
    sp[tid] = v;
  }
  if (tid < 3) {  // right pad (multiplies only zero columns of H; keep finite)
    int t = blockbase + 2048 + tid;
    float v = 0.0f;
    if (t < T) { float4 q = xr[t]; v = q.w + q.x - q.z; }
    sp[2053 + tid] = v;
  }

  const int wave = tid >> 5;
  const int lane = tid & 31;
  const int m  = lane & 15;   // A-row / B-column / D-column index
  const int hi = lane >> 4;
  const int woff = wave * 256;

  // Fetch constant A fragments (6 x b128 loads, overlapped with staging).
  v2f a1[6], a2[6];
#pragma unroll
  for (int c = 0; c < 6; ++c) {
    float4 av = *(const float4*)&A_TAB.v[c][lane][0];
    a1[c][0] = av.x; a1[c][1] = av.y;
    a2[c][0] = av.z; a2[c][1] = av.w;
  }

  __syncthreads();

  v8f acc1 = {};
  v8f acc2 = {};
  const int pbase = woff + 16 * m + 2 * hi;  // even -> 8B-aligned LDS reads
#pragma unroll
  for (int c = 0; c < 6; ++c) {
    v2f b = *(const v2f*)&sp[pbase + 4 * c];
    acc1 = __builtin_amdgcn_wmma_f32_16x16x4_f32(false, a1[c], false, b,
                                                 (short)0, acc1, false, false);
    acc2 = __builtin_amdgcn_wmma_f32_16x16x4_f32(false, a2[c], false, b,
                                                 (short)0, acc2, false, false);
  }

  // D layout: VGPR d -> element u[tile + 16*m + 8*hi + d]  (contiguous 8)
  const int ob = blockbase + woff + 16 * m + 8 * hi;
  if (ob + 7 < T) {
    *(float4*)&u1[ob]     = make_float4(acc1[0], acc1[1], acc1[2], acc1[3]);
    *(float4*)&u1[ob + 4] = make_float4(acc1[4], acc1[5], acc1[6], acc1[7]);
    *(float4*)&u2[ob]     = make_float4(acc2[0], acc2[1], acc2[2], acc2[3]);
    *(float4*)&u2[ob + 4] = make_float4(acc2[4], acc2[5], acc2[6], acc2[7]);
  }
}

// ---------------------------------------------------------------------------
// Kernel 2: the inherently serial reservoir recurrence (latency-bound).
// Dependent chain per step, with 1/x3 powers pre-folded:
//   gw = C1 * r^3 * sqrt(r)            (C1 = x2 * x3^-3.5)
//   rn = max(0, (r+u1) + gw)
//   z  = fma(rn^4, C2, 1)              (C2 = x3^-4)
//   tt = v_rsq(v_sqrt(z))              ( == (1+(rn/x3)^4)^-0.25 )
//   r' = rn * tt ;  q = (rn - r') + max(0, u2 + gw)
// Raw v_sqrt_f32 / v_rsq_f32 keep the chain to single-instruction trans ops;
// float4-batched I/O + global_prefetch keep memory off the chain.
// ---------------------------------------------------------------------------
__global__ void scan_kernel(const float* __restrict__ u1,
                            const float* __restrict__ u2,
                            const float* __restrict__ x2p,
                            const float* __restrict__ x3p,
                            float* __restrict__ qsim,
                            float* __restrict__ rstore, int T) {
  if (threadIdx.x != 0 || blockIdx.x != 0) return;
  const float x2 = x2p[0];
  const float x3 = x3p[0];
  const float inv_x3 = 1.0f / x3;
  const float inv2 = inv_x3 * inv_x3;
  const float C1 = x2 * inv2 * inv_x3 * __builtin_amdgcn_sqrtf(inv_x3);
  const float C2 = inv2 * inv2;
  float r = 0.0f;  // R_INIT * x3

  const float4* __restrict__ u1v = (const float4*)u1;
  const float4* __restrict__ u2v = (const float4*)u2;
  float4* __restrict__ qv = (float4*)qsim;
  float4* __restrict__ rv = (float4*)rstore;

  const int n4 = T >> 2;
  for (int i = 0; i < n4; ++i) {
    if (i + 16 < n4) {
      __builtin_prefetch(&u1v[i + 16], 0, 1);
      __builtin_prefetch(&u2v[i + 16], 0, 1);
    }
    float4 a = u1v[i];
    float4 b = u2v[i];
    float4 qo, ro;
#pragma unroll
    for (int s = 0; s < 4; ++s) {
      float ua = (s == 0) ? a.x : (s == 1) ? a.y : (s == 2) ? a.z : a.w;
      float ub = (s == 0) ? b.x : (s == 1) ? b.y : (s == 2) ? b.z : b.w;
      float sr  = __builtin_amdgcn_sqrtf(r);
      float r2  = r * r;
      float gw  = (r2 * r) * (C1 * sr);            // x2 * (r/x3)^3.5
      float rn  = fmaxf(0.0f, (r + ua) + gw);      // (r+ua) off-chain
      float rn2 = rn * rn;
      float z   = __builtin_fmaf(rn2 * rn2, C2, 1.0f);
      float tt  = __builtin_amdgcn_rsqf(__builtin_amdgcn_sqrtf(z));
      float rq  = rn * tt;                         // reservoir after release
      float qd  = fmaxf(0.0f, ub + gw);
      float qs  = (rn - rq) + qd;
      r = rq;
      if (s == 0)      { qo.x = qs; ro.x = r; }
      else if (s == 1) { qo.y = qs; ro.y = r; }
      else if (s == 2) { qo.z = qs; ro.z = r; }
      else             { qo.w = qs; ro.w = r; }
    }
    qv[i] = qo;
    rv[i] = ro;
  }
}

// ---------------------------------------------------------------------------
extern "C" void kernel_launch(void* const* d_in, const int* in_sizes, int n_in,
                              void* d_out, int out_size, void* d_ws,
                              size_t ws_size, hipStream_t stream) {
  const float* x  = (const float*)d_in[0];  // (T,4) row-major
  const float* x2 = (const float*)d_in[1];
  const float* x3 = (const float*)d_in[2];
  const int T = in_sizes[0] / 4;

  float* u1 = (float*)d_ws;        // T floats
  float* u2 = u1 + T;              // T floats
  float* qsim = (float*)d_out;     // T floats  (qsim, flattened (T,1))
  float* rstore = qsim + T;        // T floats  (r_store)

  const int nblocks = (T + 2047) / 2048;
  fir_wmma_kernel<<<nblocks, 256, 0, stream>>>(x, u1, u2, T);
  scan_kernel<<<1, 32, 0, stream>>>(u1, u2, x2, x3, qsim, rstore, T);
}